// Attention_3496103379511
// MI455X (gfx1250) — compile-verified
//
#include <hip/hip_runtime.h>
#include <math.h>

// Problem constants (match reference)
#define BB 64
#define TT 2048
#define EE 512
#define HH 128
#define DLL 1024
#define CCC 32
#define KW 31
#define TCH 8          // t-chunks for the context bmm

typedef __attribute__((ext_vector_type(2))) float v2f;
typedef __attribute__((ext_vector_type(8))) float v8f;

// Branch-free tanh on the hardware exp + rcp units (v_exp_f32, v_rcp_f32).
// tanh(x) = sign(x) * (1 - 2e/(1+e)),  e = exp(-2|x|) in (0,1]  => no inf,
// no division ladder (v_div_scale/v_div_fmas), ~5 VALU + 2 TRANS total.
__device__ __forceinline__ float fast_tanhf(float x) {
  const float a = fabsf(x);
  const float e = __expf(-2.0f * a);                 // v_exp_f32
  const float r = __builtin_amdgcn_rcpf(1.0f + e);   // v_rcp_f32
  const float t = fmaf(-2.0f * e, r, 1.0f);
  return copysignf(t, x);
}

// ---------------------------------------------------------------------------
// Kernel 1: processed_lstm  pl[b,h] = sum_d lstm[b,d] * W_lstm[h,d]
// ---------------------------------------------------------------------------
__global__ __launch_bounds__(HH) void pl_kernel(const float* __restrict__ lstm,
                                                const float* __restrict__ W_lstm,
                                                float* __restrict__ pl) {
  const int b = blockIdx.x;
  const int h = threadIdx.x;
  const float4* x = (const float4*)(lstm + (size_t)b * DLL);
  const float4* w = (const float4*)(W_lstm + (size_t)h * DLL);
  float acc = 0.0f;
  for (int d = 0; d < DLL / 4; ++d) {
    float4 xv = x[d], wv = w[d];
    acc = fmaf(xv.x, wv.x, acc);
    acc = fmaf(xv.y, wv.y, acc);
    acc = fmaf(xv.z, wv.z, acc);
    acc = fmaf(xv.w, wv.w, acc);
  }
  pl[b * HH + h] = acc;
}

// ---------------------------------------------------------------------------
// Kernel 2: fused location-conv + (loc @ W_loc^T) via V_WMMA_F32_16X16X4_F32
//           + tanh + dot with W_e  ->  energies[b,t]
// Grid: (B, T/128). Block: 256 threads = 8 waves; each wave owns 16 t-rows.
// K=32 -> 8 chained k=4 f32 WMMAs per 16x16 output tile, 8 N-tiles (H=128).
// ---------------------------------------------------------------------------
__global__ __launch_bounds__(256) void energies_kernel(
    const float* __restrict__ pe,      // [B,T,H]
    const float* __restrict__ awc,     // [B,T]
    const float* __restrict__ conv_w,  // [CC,1,KW]
    const float* __restrict__ W_loc,   // [H,CC]
    const float* __restrict__ W_e,     // [H]
    const float* __restrict__ pl,      // [B,H]
    float* __restrict__ energies)      // [B,T]
{
  __shared__ float s_wloc[HH * CCC];   // 16 KB
  __shared__ float s_conv[CCC * KW];   // ~4 KB
  __shared__ float s_pl[HH];
  __shared__ float s_we[HH];

  const int b    = blockIdx.x;
  const int tblk = blockIdx.y;         // 128 t-values per block
  const int tid  = threadIdx.x;

  for (int i = tid; i < HH * CCC; i += 256) s_wloc[i] = W_loc[i];
  for (int i = tid; i < CCC * KW; i += 256) s_conv[i] = conv_w[i];
  if (tid < HH) { s_pl[tid] = pl[b * HH + tid]; s_we[tid] = W_e[tid]; }
  __syncthreads();

  const int wave  = tid >> 5;          // 0..7
  const int lane  = tid & 31;
  const int hl    = lane & 15;         // position within half
  const int half  = lane >> 4;         // 0 or 1
  const int cpair = half * 2;          // channel-pair base within a k=4 group
  const int t0    = tblk * 128 + wave * 16;
  const int trow  = t0 + hl;

  // --- location conv window (zero-padded), 31 taps ---
  float w[KW];
  const float* awb = awc + (size_t)b * TT;
#pragma unroll
  for (int j = 0; j < KW; ++j) {
    int tt = trow - (KW / 2) + j;
    w[j] = (tt >= 0 && tt < TT) ? awb[tt] : 0.0f;
  }

  // --- A fragments: afr[2s+u] = loc[trow][4s + cpair + u]  (16 values/lane) ---
  float afr[16];
#pragma unroll
  for (int s = 0; s < 8; ++s) {
#pragma unroll
    for (int u = 0; u < 2; ++u) {
      const int c = 4 * s + cpair + u;
      const float* cw = &s_conv[c * KW];
      float acc = 0.0f;
#pragma unroll
      for (int j = 0; j < KW; ++j) acc = fmaf(cw[j], w[j], acc);
      afr[2 * s + u] = acc;
    }
  }

  // --- 8 N-tiles over H: chained f32 WMMAs, then fused tanh/W_e epilogue ---
  float part[8];
#pragma unroll
  for (int r = 0; r < 8; ++r) part[r] = 0.0f;

  const float* peb = pe + ((size_t)b * TT + (size_t)t0) * HH;
  const int rowbase = half * 8;        // D rows 0..7 (lanes 0-15) / 8..15 (16-31)

#pragma unroll 1
  for (int n = 0; n < 8; ++n) {
    const int hcol = n * 16 + hl;
    v8f acc = {};                      // C = 0
#pragma unroll
    for (int s = 0; s < 8; ++s) {
      v2f a; a.x = afr[2 * s]; a.y = afr[2 * s + 1];
      const int c = 4 * s + cpair;
      v2f bf; bf.x = s_wloc[hcol * CCC + c]; bf.y = s_wloc[hcol * CCC + c + 1];
      acc = __builtin_amdgcn_wmma_f32_16x16x4_f32(
          /*neg_a=*/false, a, /*neg_b=*/false, bf,
          /*c_mod=*/(short)0, acc, /*reuse_a=*/false, /*reuse_b=*/false);
    }
    const float we_h = s_we[hcol];
    const float pl_h = s_pl[hcol];
#pragma unroll
    for (int r = 0; r < 8; ++r) {
      const float x = acc[r] + pl_h + peb[(size_t)(rowbase + r) * HH + hcol];
      part[r] += we_h * fast_tanhf(x);
    }
  }

  // --- reduce over the 16 lanes of each half (h-dimension) ---
#pragma unroll
  for (int r = 0; r < 8; ++r) {
    float v = part[r];
    v += __shfl_xor(v, 1, 32);
    v += __shfl_xor(v, 2, 32);
    v += __shfl_xor(v, 4, 32);
    v += __shfl_xor(v, 8, 32);
    part[r] = v;
  }
  if (hl == 0) {
#pragma unroll
    for (int r = 0; r < 8; ++r)
      energies[(size_t)b * TT + t0 + rowbase + r] = part[r];
  }
}

// ---------------------------------------------------------------------------
// Kernel 3: softmax over T per batch -> attention_weights (written to d_out)
// ---------------------------------------------------------------------------
__global__ __launch_bounds__(256) void softmax_kernel(const float* __restrict__ energies,
                                                      float* __restrict__ weights) {
  __shared__ float red[256];
  const int b = blockIdx.x;
  const int tid = threadIdx.x;
  const float* e = energies + (size_t)b * TT;

  float m = -INFINITY;
  float vals[TT / 256];
#pragma unroll
  for (int k = 0; k < TT / 256; ++k) {
    vals[k] = e[k * 256 + tid];
    m = fmaxf(m, vals[k]);
  }
  red[tid] = m; __syncthreads();
  for (int s = 128; s > 0; s >>= 1) {
    if (tid < s) red[tid] = fmaxf(red[tid], red[tid + s]);
    __syncthreads();
  }
  m = red[0]; __syncthreads();

  float sum = 0.0f;
#pragma unroll
  for (int k = 0; k < TT / 256; ++k) {
    vals[k] = __expf(vals[k] - m);     // v_exp_f32
    sum += vals[k];
  }
  red[tid] = sum; __syncthreads();
  for (int s = 128; s > 0; s >>= 1) {
    if (tid < s) red[tid] += red[tid + s];
    __syncthreads();
  }
  const float inv = 1.0f / red[0];
#pragma unroll
  for (int k = 0; k < TT / 256; ++k)
    weights[(size_t)b * TT + k * 256 + tid] = vals[k] * inv;
}

// ---------------------------------------------------------------------------
// Kernel 4: context partials  part[tc][b][e] = sum_{t in chunk} w[b,t]*enc[b,t,e]
// 268 MB streaming pass -> the op's bandwidth floor. float4 lanes => b128
// loads (512 B/wave/instr); 4-deep t-unroll keeps MBs of loads in flight.
// Grid (B, TCH); block 128 threads, each owns 4 consecutive e-columns.
// ---------------------------------------------------------------------------
__global__ __launch_bounds__(128) void ctx_partial_kernel(const float* __restrict__ enc,
                                                          const float* __restrict__ wts,
                                                          float* __restrict__ part) {
  const int b  = blockIdx.x;
  const int tc = blockIdx.y;
  const int e4 = threadIdx.x;              // float4 column index, 0..127
  const int TC = TT / TCH;                 // 256 t per chunk
  const float4* encb = (const float4*)(enc + ((size_t)b * TT + (size_t)tc * TC) * EE) + e4;
  const float* wb    = wts + (size_t)b * TT + (size_t)tc * TC;

  float4 acc = {0.f, 0.f, 0.f, 0.f};
#pragma unroll 4
  for (int t = 0; t < TC; ++t) {
    const float wv = wb[t];                // block-uniform -> scalar load
    const float4 ev = encb[(size_t)t * (EE / 4)];
    acc.x = fmaf(wv, ev.x, acc.x);
    acc.y = fmaf(wv, ev.y, acc.y);
    acc.z = fmaf(wv, ev.z, acc.z);
    acc.w = fmaf(wv, ev.w, acc.w);
  }
  float4* p = (float4*)(part + ((size_t)tc * BB + b) * EE) + e4;
  *p = acc;
}

__global__ __launch_bounds__(256) void ctx_reduce_kernel(const float* __restrict__ part,
                                                         float* __restrict__ out) {
  const int idx = blockIdx.x * 256 + threadIdx.x;  // float4 index, 0 .. B*E/4-1
  const float4* p = (const float4*)part;
  float4 acc = {0.f, 0.f, 0.f, 0.f};
#pragma unroll
  for (int tc = 0; tc < TCH; ++tc) {
    float4 v = p[(size_t)tc * (BB * EE / 4) + idx];
    acc.x += v.x; acc.y += v.y; acc.z += v.z; acc.w += v.w;
  }
  ((float4*)out)[idx] = acc;
}

// ---------------------------------------------------------------------------
extern "C" void kernel_launch(void* const* d_in, const int* in_sizes, int n_in,
                              void* d_out, int out_size, void* d_ws, size_t ws_size,
                              hipStream_t stream) {
  const float* enc   = (const float*)d_in[0];  // [B,T,E]
  const float* pe    = (const float*)d_in[1];  // [B,T,H]
  const float* lstm  = (const float*)d_in[2];  // [B,1,DL]
  const float* awc   = (const float*)d_in[3];  // [B,T]
  const float* Wlstm = (const float*)d_in[4];  // [H,DL]
  const float* convw = (const float*)d_in[5];  // [CC,1,K]
  const float* Wloc  = (const float*)d_in[6];  // [H,CC]
  const float* We    = (const float*)d_in[7];  // [1,H]

  float* out = (float*)d_out;
  float* ctx_out = out;                 // [B,E]  (attention_context)
  float* wts_out = out + BB * EE;       // [B,T]  (attention_weights)

  float* ws = (float*)d_ws;
  float* pl_ws   = ws;                                   // B*H        = 8192
  float* en_ws   = pl_ws + BB * HH;                      // B*T        = 131072
  float* part_ws = en_ws + BB * TT;                      // TCH*B*E    = 262144

  pl_kernel<<<BB, HH, 0, stream>>>(lstm, Wlstm, pl_ws);
  energies_kernel<<<dim3(BB, TT / 128), 256, 0, stream>>>(pe, awc, convw, Wloc, We,
                                                          pl_ws, en_ws);
  softmax_kernel<<<BB, 256, 0, stream>>>(en_ws, wts_out);
  ctx_partial_kernel<<<dim3(BB, TCH), 128, 0, stream>>>(enc, wts_out, part_ws);
  ctx_reduce_kernel<<<(BB * EE / 4) / 256, 256, 0, stream>>>(part_ws, ctx_out);
}